// HierarchicalSparseAttention_34076270526937
// MI455X (gfx1250) — compile-verified
//
#include <hip/hip_runtime.h>

// Tree-structured parent-child attention, B=8 N=2048 H=16 D=64, fp32.
// Memory-bound (~0.4 FLOP/byte): stream everything once, NT hints,
// scores via V_WMMA_F32_16X16X4_F32 diag(Q K^T) per node (compute is free).

typedef __attribute__((ext_vector_type(2))) float v2f;
typedef __attribute__((ext_vector_type(4))) float v4f;
typedef __attribute__((ext_vector_type(8))) float v8f;

#define NHEAD 16
#define DHEAD 64
#define NODE_ELEMS (NHEAD * DHEAD)   // 1024 floats per (b,n)
#define WAVES_PER_BLOCK 8

__global__ __launch_bounds__(256, 4)
void tree_attn_wmma_kernel(const float* __restrict__ Qp,
                           const float* __restrict__ Kp,
                           const float* __restrict__ Vp,
                           const float* __restrict__ Kc,
                           const float* __restrict__ Vc,
                           float* __restrict__ Out,
                           int num_nodes)
{
    // per-wave score slab: [wave][{self,c0,c1}][head]
    __shared__ float s_sc[WAVES_PER_BLOCK][3][NHEAD];

    const int tid  = threadIdx.x;
    const int lane = tid & 31;
    const int wave = tid >> 5;
    const int hl   = lane >> 4;      // lane half (0: lanes 0-15, 1: lanes 16-31)
    const int hh   = lane & 15;

    int node = blockIdx.x * WAVES_PER_BLOCK + wave;
    if (node >= num_nodes) node = num_nodes - 1;  // tail waves redo last node (benign)

    const size_t qbase  = (size_t)node * NODE_ELEMS;                 // Q_p/K_p/V_p/Out row
    const size_t c0base = (size_t)(2 * node) * NODE_ELEMS;           // child 2n in K_c/V_c
    const size_t c1base = c0base + NODE_ELEMS;                       // child 2n+1

    // ---------------- Phase 1: scores via WMMA f32 16x16x4, K-loop over D=64 --------------
    // A (Q, 16x4):  lane<16 holds (M=hh, K=0..1), lane>=16 holds (M=hh, K=2..3)
    // B (K^T, 4x16): col N=hh in lane hh; lanes 0-15 hold K=0..1, lanes 16-31 K=2..3
    // -> both operands are the same float2 fetch: base + hh*64 + 4t + 2*hl
    const int loff = hh * DHEAD + 2 * hl;

    v8f cs = {0.f,0.f,0.f,0.f,0.f,0.f,0.f,0.f};
    v8f c0 = cs, c1 = cs;

#pragma unroll
    for (int t = 0; t < DHEAD / 4; ++t) {
        const int o = loff + 4 * t;
        v2f a  = __builtin_nontemporal_load((const v2f*)(Qp + qbase  + o));
        v2f bs = __builtin_nontemporal_load((const v2f*)(Kp + qbase  + o));
        v2f b0 = __builtin_nontemporal_load((const v2f*)(Kc + c0base + o));
        v2f b1 = __builtin_nontemporal_load((const v2f*)(Kc + c1base + o));
        cs = __builtin_amdgcn_wmma_f32_16x16x4_f32(false, a, false, bs, (short)0, cs, false, false);
        c0 = __builtin_amdgcn_wmma_f32_16x16x4_f32(false, a, false, b0, (short)0, c0, false, false);
        c1 = __builtin_amdgcn_wmma_f32_16x16x4_f32(false, a, false, b1, (short)0, c1, false, false);
    }

    // Diagonal of C (doc layout): head r   -> lane r,    component r   (r = 0..7)
    //                             head 8+r -> lane 24+r, component r
#pragma unroll
    for (int r = 0; r < 8; ++r) {
        if (lane == r) {
            s_sc[wave][0][r] = cs[r];
            s_sc[wave][1][r] = c0[r];
            s_sc[wave][2][r] = c1[r];
        }
        if (lane == 24 + r) {
            s_sc[wave][0][8 + r] = cs[r];
            s_sc[wave][1][8 + r] = c0[r];
            s_sc[wave][2][8 + r] = c1[r];
        }
    }
    __syncthreads();

    // ---------------- Phase 2: 3-way softmax + weighted V blend ---------------------------
    const float sm_scale = 0.125f;  // 1/sqrt(64)
#pragma unroll
    for (int i = 0; i < NHEAD / 2; ++i) {
        const int h = 2 * i + hl;                      // half-wave per head, 16 lanes x float4
        float ss = s_sc[wave][0][h] * sm_scale;
        float s0 = s_sc[wave][1][h] * sm_scale;
        float s1 = s_sc[wave][2][h] * sm_scale;
        float m  = fmaxf(ss, fmaxf(s0, s1));
        float es = __expf(ss - m);
        float e0 = __expf(s0 - m);
        float e1 = __expf(s1 - m);
        float inv = 1.0f / (es + e0 + e1 + 1e-9f);
        float ws = es * inv, w0 = e0 * inv, w1 = e1 * inv;

        const size_t vo = (size_t)h * DHEAD + hh * 4;
        v4f vp  = __builtin_nontemporal_load((const v4f*)(Vp + qbase  + vo));
        v4f vc0 = __builtin_nontemporal_load((const v4f*)(Vc + c0base + vo));
        v4f vc1 = __builtin_nontemporal_load((const v4f*)(Vc + c1base + vo));
        v4f o   = ws * vp + w0 * vc0 + w1 * vc1;
        __builtin_nontemporal_store(o, (v4f*)(Out + qbase + vo));
    }
}

extern "C" void kernel_launch(void* const* d_in, const int* in_sizes, int n_in,
                              void* d_out, int out_size, void* d_ws, size_t ws_size,
                              hipStream_t stream)
{
    const float* Qp = (const float*)d_in[0];
    const float* Kp = (const float*)d_in[1];
    const float* Vp = (const float*)d_in[2];
    const float* Kc = (const float*)d_in[3];
    const float* Vc = (const float*)d_in[4];
    float* Out = (float*)d_out;

    const int num_nodes = in_sizes[0] / NODE_ELEMS;   // B*N = 16384
    const int blocks = (num_nodes + WAVES_PER_BLOCK - 1) / WAVES_PER_BLOCK;

    tree_attn_wmma_kernel<<<dim3(blocks), dim3(256), 0, stream>>>(
        Qp, Kp, Vp, Kc, Vc, Out, num_nodes);
}